// ChunkedDensePairwiseRelationModuleOptimized_81415400063724
// MI455X (gfx1250) — compile-verified
//
#include <hip/hip_runtime.h>
#include <hip/hip_bf16.h>
#include <stdint.h>

// Problem constants (from reference): B=4, N=256, D=320, L=256, H=256
#define BB 4
#define NN 256
#define DD 320
#define LL 256
#define HH 256

typedef float v2f __attribute__((ext_vector_type(2)));
typedef float v8f __attribute__((ext_vector_type(8)));

__device__ __forceinline__ v8f wmma_f32(v2f a, v2f b, v8f c) {
    // V_WMMA_F32_16X16X4_F32 (neg_a, A, neg_b, B, c_mod, C, reuse_a, reuse_b)
    // f32 NEG bits are C-only per ISA, so neg_a/neg_b stay false.
    return __builtin_amdgcn_wmma_f32_16x16x4_f32(false, a, false, b, (short)0, c,
                                                 false, false);
}

// ---------------------------------------------------------------------------
// Kernel A: build ai/aj  (one wave = one 16x16 output tile for BOTH matrices)
//   ai[b,i,k] = feat_i . Wi  + p_i . Wg + lang_b . Wl + b1[k]
//   aj[b,j,k] = feat_j . Wj  - p_j . Wg
// All via V_WMMA_F32_16X16X4_F32, K-loop in steps of 4. Geo step is fully
// branch-free: all loads are unconditional (clamped offsets within a single
// base), selection happens on values (v_cndmask), and the K=6,7 zero padding
// lives only in the A operand (zero A rows make the B values don't-care).
// ---------------------------------------------------------------------------
__global__ void prep_kernel(const float* __restrict__ feat,
                            const float* __restrict__ lang,
                            const float* __restrict__ centers,
                            const float* __restrict__ sizes,
                            const float* __restrict__ w1,
                            const float* __restrict__ b1,
                            float* __restrict__ ai,
                            float* __restrict__ aj) {
    const int m0   = blockIdx.x * 16;   // row tile over B*N (never straddles a batch)
    const int n0   = blockIdx.y * 16;   // col tile over H
    const int lane = threadIdx.x;       // 0..31 (wave32)
    const int half = lane >> 4;
    const int rl   = lane & 15;         // A-row / B-col within tile
    const int kp   = half * 2;          // K sub-pair: lanes 0-15 -> K0/K1, 16-31 -> K2/K3
    const int b    = m0 >> 8;           // batch index

    const float* wi_base = w1;                       // rows [0, 320)
    const float* wj_base = w1 + DD * HH;             // rows [320, 640)
    const float* wg_base = w1 + 2 * DD * HH;         // rows [640, 646)
    const float* wl_base = w1 + (2 * DD + 6) * HH;   // rows [646, 902)
    const int col = n0 + rl;

    v8f acc_i = {};
    v8f acc_j = {};

    // --- object features: K = 320, shared A operand for the Wi / Wj GEMMs ---
    const float* frow = feat + (m0 + rl) * DD;
    for (int kk = 0; kk < DD; kk += 4) {
        const int k0 = kk + kp;
        v2f a;
        a.x = frow[k0];
        a.y = frow[k0 + 1];
        v2f bi;
        bi.x = wi_base[(k0)     * HH + col];
        bi.y = wi_base[(k0 + 1) * HH + col];
        v2f bj;
        bj.x = wj_base[(k0)     * HH + col];
        bj.y = wj_base[(k0 + 1) * HH + col];
        acc_i = wmma_f32(a, bi, acc_i);
        acc_j = wmma_f32(a, bj, acc_j);
    }

    // --- geometry: K = 6 padded to 8, p = [c/5, s/2], aj subtracts (negated B).
    {
        const float* crow = centers + (m0 + rl) * 3;
        const float* srow = sizes   + (m0 + rl) * 3;

        // Unconditional loads; selection on values only (no divergent branches).
        const float c_kp = crow[kp];        // g = 0 (half 0) / g = 2 (half 1): centers
        const float c1   = crow[1];         // candidate for g = 1
        const float s0   = srow[0];         // candidate for g = 3
        const float s1v  = srow[1];         // g = 4
        const float s2v  = srow[2];         // g = 5

        // step 0: K rows {0,1} (lanes 0-15) / {2,3} (lanes 16-31) -- all valid.
        {
            v2f a;
            a.x = c_kp * 0.2f;
            a.y = half ? (s0 * 0.5f) : (c1 * 0.2f);
            v2f bg;
            bg.x = wg_base[(kp)     * HH + col];
            bg.y = wg_base[(kp + 1) * HH + col];
            v2f bgn; bgn.x = -bg.x; bgn.y = -bg.y;
            acc_i = wmma_f32(a, bg,  acc_i);
            acc_j = wmma_f32(a, bgn, acc_j);
        }
        // step 1: K rows {4,5} (lanes 0-15) / {6,7} pad (lanes 16-31).
        // A rows 6,7 are zero, so the (clamped-address) B values there are
        // don't-care: zero contribution without any B-side masking.
        {
            v2f a;
            a.x = half ? 0.0f : (s1v * 0.5f);
            a.y = half ? 0.0f : (s2v * 0.5f);
            const int r0 = half ? 5 : 4;    // clamped wg row (same base, cndmask offset)
            const int r1 = 5;
            v2f bg;
            bg.x = wg_base[r0 * HH + col];
            bg.y = wg_base[r1 * HH + col];
            v2f bgn; bgn.x = -bg.x; bgn.y = -bg.y;
            acc_i = wmma_f32(a, bg,  acc_i);
            acc_j = wmma_f32(a, bgn, acc_j);
        }
    }

    // --- language: K = 256, broadcast row (folds fl into ai) ---
    const float* lrow = lang + b * LL;
    for (int kk = 0; kk < LL; kk += 4) {
        const int k0 = kk + kp;
        v2f a;
        a.x = lrow[k0];
        a.y = lrow[k0 + 1];
        v2f bl;
        bl.x = wl_base[(k0)     * HH + col];
        bl.y = wl_base[(k0 + 1) * HH + col];
        acc_i = wmma_f32(a, bl, acc_i);
    }

    // Epilogue: C layout VGPR r -> M = r + 8*half, N = lane&15. Fold b1 into ai.
    const float b1v = b1[col];
#pragma unroll
    for (int r = 0; r < 8; ++r) {
        const int row = m0 + r + 8 * half;
        ai[row * HH + col] = acc_i[r] + b1v;
        aj[row * HH + col] = acc_j[r];
    }
}

// ---------------------------------------------------------------------------
// Kernel B: fused pairwise score + mask + softmax + rel_w write.
// One block per (b,i); thread t owns column j=t. ai row + w2 in LDS,
// aj streamed as float4 with global_prefetch_b8 ahead of the stream.
// ---------------------------------------------------------------------------
__global__ void score_softmax_kernel(const float* __restrict__ ai,
                                     const float* __restrict__ aj,
                                     const float* __restrict__ w2,
                                     const float* __restrict__ b2,
                                     const uint8_t* __restrict__ mask,
                                     float* __restrict__ relw) {
    __shared__ float ai_s[HH];
    __shared__ float w2_s[HH];
    __shared__ float red[NN];

    const int gr = blockIdx.x;   // b*N + i
    const int b  = gr >> 8;
    const int i  = gr & 255;
    const int t  = threadIdx.x;  // = j

    ai_s[t] = ai[gr * HH + t];
    w2_s[t] = w2[t];
    __syncthreads();

    const float4* ajrow = (const float4*)(aj + (b * NN + t) * HH);
    float acc = 0.0f;
#pragma unroll 4
    for (int kc = 0; kc < HH / 4; ++kc) {
        if ((kc & 15) == 0) __builtin_prefetch(ajrow + kc + 16, 0, 1);
        const float4 v = ajrow[kc];
        const int k = kc * 4;
        acc += fmaxf(ai_s[k + 0] + v.x, 0.0f) * w2_s[k + 0];
        acc += fmaxf(ai_s[k + 1] + v.y, 0.0f) * w2_s[k + 1];
        acc += fmaxf(ai_s[k + 2] + v.z, 0.0f) * w2_s[k + 2];
        acc += fmaxf(ai_s[k + 3] + v.w, 0.0f) * w2_s[k + 3];
    }
    float score = acc + b2[0];
    const bool valid = mask[b * NN + i] && mask[b * NN + t];
    score = valid ? score : -1.0e9f;

    // block-wide max
    red[t] = score;
    __syncthreads();
    for (int s = NN / 2; s > 0; s >>= 1) {
        if (t < s) red[t] = fmaxf(red[t], red[t + s]);
        __syncthreads();
    }
    const float mx = red[0];
    __syncthreads();

    const float e = __expf(score - mx);
    red[t] = e;
    __syncthreads();
    for (int s = NN / 2; s > 0; s >>= 1) {
        if (t < s) red[t] += red[t + s];
        __syncthreads();
    }
    const float inv = 1.0f / red[0];

    relw[gr * NN + t] = e * inv;
}

// ---------------------------------------------------------------------------
// Kernel C: ctx = rel_w @ feat, fused residual add.  f32 WMMA, one wave per
// 16x16 tile, K = 256 (over j).
// ---------------------------------------------------------------------------
__global__ void ctx_kernel(const float* __restrict__ relw,
                           const float* __restrict__ feat,
                           float* __restrict__ out_enh) {
    const int m0   = blockIdx.x * 16;  // i tile
    const int n0   = blockIdx.y * 16;  // d tile
    const int b    = blockIdx.z;
    const int lane = threadIdx.x;
    const int half = lane >> 4;
    const int rl   = lane & 15;
    const int kp   = half * 2;

    const float* wrow  = relw + (b * NN + m0 + rl) * NN;
    const float* fbase = feat + b * NN * DD + n0 + rl;

    v8f acc = {};
    for (int kk = 0; kk < NN; kk += 4) {
        const int k0 = kk + kp;
        v2f a;
        a.x = wrow[k0];
        a.y = wrow[k0 + 1];
        v2f bm;
        bm.x = fbase[(k0)     * DD];
        bm.y = fbase[(k0 + 1) * DD];
        acc = wmma_f32(a, bm, acc);
    }
#pragma unroll
    for (int r = 0; r < 8; ++r) {
        const int row = m0 + r + 8 * half;
        const int idx = (b * NN + row) * DD + n0 + rl;
        out_enh[idx] = acc[r] + feat[idx];
    }
}

// ---------------------------------------------------------------------------
extern "C" void kernel_launch(void* const* d_in, const int* in_sizes, int n_in,
                              void* d_out, int out_size, void* d_ws, size_t ws_size,
                              hipStream_t stream) {
    const float*   feat    = (const float*)d_in[0];
    const float*   lang    = (const float*)d_in[1];
    const float*   centers = (const float*)d_in[2];
    const float*   sizes   = (const float*)d_in[3];
    const uint8_t* mask    = (const uint8_t*)d_in[4];
    const float*   w1      = (const float*)d_in[5];
    const float*   b1      = (const float*)d_in[6];
    const float*   w2      = (const float*)d_in[7];
    const float*   b2      = (const float*)d_in[8];

    float* out_enh  = (float*)d_out;                // (B,N,D)
    float* out_relw = (float*)d_out + BB * NN * DD; // (B,N,N)

    float* ai = (float*)d_ws;                       // B*N*H floats (1 MB)
    float* aj = ai + BB * NN * HH;                  // B*N*H floats (1 MB)

    prep_kernel<<<dim3((BB * NN) / 16, HH / 16), 32, 0, stream>>>(
        feat, lang, centers, sizes, w1, b1, ai, aj);

    score_softmax_kernel<<<BB * NN, NN, 0, stream>>>(
        ai, aj, w2, b2, mask, out_relw);

    ctx_kernel<<<dim3(NN / 16, DD / 16, BB), 32, 0, stream>>>(
        out_relw, feat, out_enh);
}